// FFMCell_2241972928685
// MI455X (gfx1250) — compile-verified
//
#include <hip/hip_runtime.h>

// FFM cell combine: new_state = state * exp((-|a| + i*b)*j) + x
// T=4096 slices of TRACE(64) x CTX(64). Pure HBM-streaming kernel:
//   24 B/elem, ~402 MB total => ~17 us floor at 23.3 TB/s.
// Strategy:
//   * one workgroup (256 thr = 8 wave32) per t-slice
//   * hoist exp(-|a[m]|*t) and cos/sin(b[c]*t) into LDS (64x trans reduction)
//   * b128 vectorized, non-temporal global traffic (stream > 192MB L2)
// WMMA intentionally unused: elementwise complex FMA has no K-contraction.

#define TRACE   64
#define CTX     64
#define SLICE   (TRACE * CTX)        // 4096 elements per t
#define THREADS 256
#define VECS    (SLICE / 4)          // 1024 float4 per slice
#define KITERS  (VECS / THREADS)     // 4 float4 per thread per array

typedef float v4f __attribute__((ext_vector_type(4)));

__global__ __launch_bounds__(THREADS) void ffm_combine_kernel(
    const float* __restrict__ a,        // [TRACE]
    const float* __restrict__ b,        // [CTX]
    const float* __restrict__ state_re, // [T*TRACE*CTX]
    const float* __restrict__ state_im,
    const float* __restrict__ x_re,
    const float* __restrict__ x_im,
    const int*  __restrict__ iv,        // [T]
    const int*  __restrict__ jv,        // [T]
    float* __restrict__ out_re,
    float* __restrict__ out_im,
    float* __restrict__ out_ij)         // [T] = float(j+i)
{
    __shared__ __align__(16) float s_decay[TRACE]; // exp(-|a[m]|*t)
    __shared__ __align__(16) float s_cr[CTX];      // cos(b[c]*t)
    __shared__ __align__(16) float s_ci[CTX];      // sin(b[c]*t)

    const int t   = blockIdx.x;
    const int tid = threadIdx.x;
    const float tf = (float)jv[t];

    // Phase 1: per-slice transcendental tables (trans ops: 3*64 per 4096 elems)
    if (tid < CTX) {
        const float ang = b[tid] * tf;
        s_cr[tid] = __cosf(ang);
        s_ci[tid] = __sinf(ang);
    } else if (tid < CTX + TRACE) {
        const int m = tid - CTX;
        s_decay[m] = __expf(-fabsf(a[m]) * tf);
    }
    if (tid == 0) out_ij[t] = (float)(jv[t] + iv[t]);
    __syncthreads();

    // Phase 2: stream the slice with b128 non-temporal vmem
    const size_t base = (size_t)t * SLICE;
    const v4f* __restrict__ sre = (const v4f*)(state_re + base);
    const v4f* __restrict__ sim = (const v4f*)(state_im + base);
    const v4f* __restrict__ xre = (const v4f*)(x_re     + base);
    const v4f* __restrict__ xim = (const v4f*)(x_im     + base);
    v4f* __restrict__ ore = (v4f*)(out_re + base);
    v4f* __restrict__ oim = (v4f*)(out_im + base);

#pragma unroll
    for (int k = 0; k < KITERS; ++k) {
        const int v  = tid + k * THREADS;   // float4 index in slice
        const int e  = v << 2;              // element index
        const int m  = e >> 6;              // row   (TRACE)
        const int c0 = e & (CTX - 1);       // col 0 (multiple of 4)

        const float d  = s_decay[m];                  // ds broadcast
        const v4f   cr = *(const v4f*)&s_cr[c0];      // ds_load_b128
        const v4f   ci = *(const v4f*)&s_ci[c0];
        const v4f   gr = cr * d;                      // gamma
        const v4f   gi = ci * d;

        const v4f sr = __builtin_nontemporal_load(sre + v);
        const v4f si = __builtin_nontemporal_load(sim + v);
        const v4f xr = __builtin_nontemporal_load(xre + v);
        const v4f xi = __builtin_nontemporal_load(xim + v);

        const v4f rr = sr * gr - si * gi + xr;        // Re(state*gamma)+Re(x)
        const v4f ri = sr * gi + si * gr + xi;        // Im(state*gamma)+Im(x)

        __builtin_nontemporal_store(rr, ore + v);
        __builtin_nontemporal_store(ri, oim + v);
    }
}

extern "C" void kernel_launch(void* const* d_in, const int* in_sizes, int n_in,
                              void* d_out, int out_size, void* d_ws, size_t ws_size,
                              hipStream_t stream) {
    (void)n_in; (void)out_size; (void)d_ws; (void)ws_size;

    const float* a   = (const float*)d_in[0];
    const float* b   = (const float*)d_in[1];
    const float* sre = (const float*)d_in[2];
    const float* sim = (const float*)d_in[3];
    const float* xre = (const float*)d_in[4];
    const float* xim = (const float*)d_in[5];
    const int*   iv  = (const int*)d_in[6];
    const int*   jv  = (const int*)d_in[7];

    const size_t N = (size_t)in_sizes[2];   // T*TRACE*CTX = 16,777,216
    const int    T = in_sizes[6];           // 4096

    float* out_re = (float*)d_out;
    float* out_im = out_re + N;
    float* out_ij = out_im + N;             // j+i, stored as f32

    ffm_combine_kernel<<<dim3(T), dim3(THREADS), 0, stream>>>(
        a, b, sre, sim, xre, xim, iv, jv, out_re, out_im, out_ij);
}